// Text2Vec_72773925863670
// MI455X (gfx1250) — compile-verified
//
#include <hip/hip_runtime.h>
#include <cstdint>

typedef float v2f __attribute__((ext_vector_type(2)));
typedef float v8f __attribute__((ext_vector_type(8)));

#define DD      64
#define EST     68            // padded LDS row stride (floats): conflict-free A loads
#define THREADS 128           // 4 wave32 -> 4 column tiles of one 16-row slab

__global__ __launch_bounds__(THREADS, 1)
void text2vec_scan_kernel(const int* __restrict__ text_ids,
                          const float* __restrict__ emb,
                          float* __restrict__ out, int S) {
  // Double-buffered 16x64 ctx slab (rows [ti, ti+16) of this batch's ctx).
  __shared__ __align__(16) float cs[2][16 * EST];

  const int blk  = blockIdx.x;
  const int b    = blk >> 2;            // batch
  const int ti   = (blk & 3) * 16;      // global row base of this slab
  const int tid  = threadIdx.x;
  const int lane = tid & 31;
  const int wave = tid >> 5;            // 0..3 -> output tile column
  const int tj   = wave * 16;
  const int ln   = lane & 15;
  const int half = lane >> 4;
  const int k2h  = 2 * half;            // lane-half selects K pair (ISA A/B layout)

  const int* ids = text_ids + (size_t)b * S;

  // ---- identity init of the slab: 1.0 at (r, ti + r) ----
  for (int i = tid; i < 16 * DD; i += THREADS) {
    const int r = i >> 6, c = i & 63;
    cs[0][r * EST + c] = (c == ti + r) ? 1.0f : 0.0f;
  }

  // Per-lane base into E for this wave's B columns: row-major [k][n], n = tj+ln.
  auto gbase = [&](int id) -> const float* {
    return emb + (size_t)id * (DD * DD) + tj + ln;
  };
  // B fragment: VGPR0 = rows {k0, k0+2} (lane halves), VGPR1 = rows {k0+1, k0+3};
  // k2h folds the half selection. Each load is a coalesced 2x64B row segment.
  auto loadB = [&](const float* base, v2f (&bf)[16]) {
    #pragma unroll
    for (int kk = 0; kk < 16; ++kk) {
      bf[kk].x = base[(kk * 4 + k2h    ) * DD];
      bf[kk].y = base[(kk * 4 + k2h + 1) * DD];
    }
  };
  // One scan step: this wave's 16x16 tile of cos(slab @ E).
  // Four interleaved WMMA accumulator chains -> dependent spacing of 4 issues,
  // hiding the XDL accumulate latency.
  auto stepc = [&](const float* crd, float* cwr, const v2f (&bf)[16]) {
    v8f ac0 = {}, ac1 = {}, ac2 = {}, ac3 = {};
    #pragma unroll
    for (int kk = 0; kk < 16; kk += 4) {
      v2f a0 = *(const v2f*)&crd[ln * EST + (kk + 0) * 4 + k2h];
      v2f a1 = *(const v2f*)&crd[ln * EST + (kk + 1) * 4 + k2h];
      v2f a2 = *(const v2f*)&crd[ln * EST + (kk + 2) * 4 + k2h];
      v2f a3 = *(const v2f*)&crd[ln * EST + (kk + 3) * 4 + k2h];
      ac0 = __builtin_amdgcn_wmma_f32_16x16x4_f32(false, a0, false, bf[kk + 0],
                                                  (short)0, ac0, false, false);
      ac1 = __builtin_amdgcn_wmma_f32_16x16x4_f32(false, a1, false, bf[kk + 1],
                                                  (short)0, ac1, false, false);
      ac2 = __builtin_amdgcn_wmma_f32_16x16x4_f32(false, a2, false, bf[kk + 2],
                                                  (short)0, ac2, false, false);
      ac3 = __builtin_amdgcn_wmma_f32_16x16x4_f32(false, a3, false, bf[kk + 3],
                                                  (short)0, ac3, false, false);
    }
    #pragma unroll
    for (int v = 0; v < 8; ++v)        // C/D layout: VGPR v -> rows v / v+8
      cwr[(8 * half + v) * EST + tj + ln] =
          __cosf((ac0[v] + ac1[v]) + (ac2[v] + ac3[v]));
  };

  // ---- software pipeline: B fragments prefetched 2 steps ahead, 3 banks ----
  v2f b0[16], b1[16], b2[16];
  loadB(gbase(ids[0]), b0);
  if (S > 1) loadB(gbase(ids[1]), b1);
  int id_q = (S > 2) ? ids[2] : 0;     // id for step t+2 (scalar, loaded early)

  __syncthreads();                      // identity slab visible

  float* csrc = cs[0];
  float* cdst = cs[1];
  int t = 0;
  while (t < S) {
    // phase 0: compute with b0, prefetch E[t+2] -> b2
    if (t + 2 < S) loadB(gbase(id_q), b2);
    id_q = (t + 3 < S) ? ids[t + 3] : 0;
    stepc(csrc, cdst, b0);
    __syncthreads();
    { float* tmp = csrc; csrc = cdst; cdst = tmp; }
    if (++t >= S) break;

    // phase 1: compute with b1, prefetch E[t+2] -> b0
    if (t + 2 < S) loadB(gbase(id_q), b0);
    id_q = (t + 3 < S) ? ids[t + 3] : 0;
    stepc(csrc, cdst, b1);
    __syncthreads();
    { float* tmp = csrc; csrc = cdst; cdst = tmp; }
    if (++t >= S) break;

    // phase 2: compute with b2, prefetch E[t+2] -> b1
    if (t + 2 < S) loadB(gbase(id_q), b1);
    id_q = (t + 3 < S) ? ids[t + 3] : 0;
    stepc(csrc, cdst, b2);
    __syncthreads();
    { float* tmp = csrc; csrc = cdst; cdst = tmp; }
    ++t;
  }

  // ---- write this slab's 16x64 final rows ----
  float* ob = out + (size_t)b * (DD * DD) + (size_t)ti * DD;
  for (int i = tid; i < 16 * DD; i += THREADS)
    ob[i] = csrc[(i >> 6) * EST + (i & 63)];
}

extern "C" void kernel_launch(void* const* d_in, const int* in_sizes, int n_in,
                              void* d_out, int out_size, void* d_ws, size_t ws_size,
                              hipStream_t stream) {
  (void)n_in; (void)d_ws; (void)ws_size;
  const int*   text_ids = (const int*)d_in[0];    // [B, S] int32
  const float* emb      = (const float*)d_in[1];  // [V, 64*64] f32
  float*       out      = (float*)d_out;          // [B, 64, 64] f32

  const int B = out_size / (DD * DD);
  const int S = in_sizes[0] / (B > 0 ? B : 1);

  text2vec_scan_kernel<<<dim3(B * 4), dim3(THREADS), 0, stream>>>(text_ids, emb, out, S);
}